// GNN_4020089389576
// MI455X (gfx1250) — compile-verified
//
#include <hip/hip_runtime.h>

typedef float v2f __attribute__((ext_vector_type(2)));
typedef float v8f __attribute__((ext_vector_type(8)));

#define NNODES 100000
#define EEDGES 1600000
#define FDIM   64
#define NGRAPH 512
#define WSTRIDE 66   // transposed-W LDS stride: pad 2 keeps 8B alignment, kills bank conflicts

// ---------------- degree / normalization ----------------
__global__ void k_deg_init(float* __restrict__ deg, int n) {
  int i = blockIdx.x * blockDim.x + threadIdx.x;
  if (i < n) deg[i] = 1.0f;  // self-loop contributes 1
}

__global__ void k_deg_count(const int* __restrict__ dst, float* __restrict__ deg, int e) {
  int i = blockIdx.x * blockDim.x + threadIdx.x;
  if (i < e) atomicAdd(&deg[dst[i]], 1.0f);
}

__global__ void k_dinv(float* __restrict__ deg, int n) {
  int i = blockIdx.x * blockDim.x + threadIdx.x;
  if (i < n) {
    float d = deg[i];
    deg[i] = (d > 0.0f) ? rsqrtf(d) : 0.0f;
  }
}

// ---------------- GEMM: t[i,:] = (relu?)in[i,:] @ W, scaled by dinv[i] ----------------
// One wave computes a 16-row x 64-col tile via 64 x V_WMMA_F32_16X16X4_F32.
// nrows % 16 == 0 (100000 = 6250*16) -> no row guards.
// A: one unconditional float2 (global_load_b64) per kt per lane.
// B: W stored TRANSPOSED in LDS (Wl[col*66+k]) so each fragment pair
//    {W[k0][col], W[k0+1][col]} is one contiguous ds_load_b64.
template <int RELU>
__global__ __launch_bounds__(256)
void k_gemm_scaled(const float* __restrict__ in, const float* __restrict__ W,
                   const float* __restrict__ dinv, float* __restrict__ out,
                   int ntiles) {
  __shared__ float Wl[FDIM * WSTRIDE];
  for (int j = threadIdx.x; j < FDIM * FDIM; j += 256) {
    const int k = j >> 6;      // W row (K index)
    const int c = j & 63;      // W col (N index)
    Wl[c * WSTRIDE + k] = W[j];
  }
  __syncthreads();

  const int wave = threadIdx.x >> 5;
  const int lane = threadIdx.x & 31;
  const int tile = blockIdx.x * 8 + wave;
  if (tile >= ntiles) return;
  const int row0 = tile * 16;

  const int lm = lane & 15;           // A: row-in-tile; B/C: col-in-ntile
  const int kh = (lane >> 4) << 1;    // 0 (lanes 0-15: K,K+1) or 2 (lanes 16-31: K+2,K+3)
  const int arow = row0 + lm;

  const float* aptr = in + (size_t)arow * FDIM + kh;
  const float* w0 = &Wl[( 0 + lm) * WSTRIDE + kh];
  const float* w1 = &Wl[(16 + lm) * WSTRIDE + kh];
  const float* w2 = &Wl[(32 + lm) * WSTRIDE + kh];
  const float* w3 = &Wl[(48 + lm) * WSTRIDE + kh];

  v8f acc0 = {}, acc1 = {}, acc2 = {}, acc3 = {};

#pragma unroll
  for (int kt = 0; kt < 16; ++kt) {
    v2f a = *(const v2f*)(aptr + kt * 4);
    if (RELU) { a.x = fmaxf(a.x, 0.0f); a.y = fmaxf(a.y, 0.0f); }
    const v2f b0 = *(const v2f*)(w0 + kt * 4);
    const v2f b1 = *(const v2f*)(w1 + kt * 4);
    const v2f b2 = *(const v2f*)(w2 + kt * 4);
    const v2f b3 = *(const v2f*)(w3 + kt * 4);
    acc0 = __builtin_amdgcn_wmma_f32_16x16x4_f32(false, a, false, b0, (short)0, acc0, false, false);
    acc1 = __builtin_amdgcn_wmma_f32_16x16x4_f32(false, a, false, b1, (short)0, acc1, false, false);
    acc2 = __builtin_amdgcn_wmma_f32_16x16x4_f32(false, a, false, b2, (short)0, acc2, false, false);
    acc3 = __builtin_amdgcn_wmma_f32_16x16x4_f32(false, a, false, b3, (short)0, acc3, false, false);
  }

  // C layout: VGPR r -> lanes 0-15: (M=r, N=lm); lanes 16-31: (M=r+8, N=lm)
  const int rofs = (lane >> 4) << 3;
#pragma unroll
  for (int r = 0; r < 8; ++r) {
    const int row = row0 + r + rofs;
    const float s = dinv[row];  // fold dinv[src] into t
    float* op = out + (size_t)row * FDIM + lm;
    op[ 0] = acc0[r] * s;
    op[16] = acc1[r] * s;
    op[32] = acc2[r] * s;
    op[48] = acc3[r] * s;
  }
}

// ---------------- self-loop + bias init: out = b + t*dinv ----------------
// (t already carries one dinv factor, so t*dinv == t_raw*dinv^2)
__global__ void k_selfinit(const float4* __restrict__ t4, const float* __restrict__ dinv,
                           const float4* __restrict__ bias4, float4* __restrict__ out4, int n) {
  int i = blockIdx.x * blockDim.x + threadIdx.x;  // over n*16 float4s
  if (i < n * 16) {
    const int node = i >> 4;
    const int f4 = i & 15;
    const float s = dinv[node];
    const float4 t = t4[i];
    const float4 b = bias4[f4];
    float4 o;
    o.x = b.x + t.x * s;
    o.y = b.y + t.y * s;
    o.z = b.z + t.z * s;
    o.w = b.w + t.w * s;
    out4[i] = o;
  }
}

// ---------------- edge scatter: out[dst] += t[src] * dinv[dst] ----------------
// One wave per edge; each lane handles 2 consecutive floats (64 = 32 lanes x 2).
__global__ __launch_bounds__(256)
void k_scatter(const int* __restrict__ src, const int* __restrict__ dst,
               const float* __restrict__ t, const float* __restrict__ dinv,
               float* __restrict__ out, int e) {
  const int wave = threadIdx.x >> 5;
  const int lane = threadIdx.x & 31;
  const int edge = blockIdx.x * 8 + wave;   // wave-uniform
  if (edge >= e) return;
  const int s = src[edge];
  const int d = dst[edge];
  const float w = dinv[d];
  const float2 v = ((const float2*)(t + (size_t)s * FDIM))[lane];
  float* op = out + (size_t)d * FDIM + lane * 2;
  atomicAdd(op + 0, v.x * w);
  atomicAdd(op + 1, v.y * w);
}

// ---------------- pooling ----------------
__global__ void k_pool_zero(float* __restrict__ pool, float* __restrict__ cnt) {
  int i = blockIdx.x * blockDim.x + threadIdx.x;
  if (i < NGRAPH * FDIM) pool[i] = 0.0f;
  if (i < NGRAPH) cnt[i] = 0.0f;
}

__global__ __launch_bounds__(256)
void k_pool_acc(const float* __restrict__ emb, const int* __restrict__ batch,
                float* __restrict__ pool, float* __restrict__ cnt, int n) {
  const int wave = threadIdx.x >> 5;
  const int lane = threadIdx.x & 31;
  const int node = blockIdx.x * 8 + wave;
  if (node >= n) return;
  const int g = batch[node];
  const float2 v = ((const float2*)(emb + (size_t)node * FDIM))[lane];
  float* pp = pool + g * FDIM + lane * 2;
  atomicAdd(pp + 0, v.x);
  atomicAdd(pp + 1, v.y);
  if (lane == 0) atomicAdd(&cnt[g], 1.0f);
}

__global__ void k_pool_out(const float* __restrict__ pool, const float* __restrict__ cnt,
                           float* __restrict__ out) {
  int i = blockIdx.x * blockDim.x + threadIdx.x;
  if (i < NGRAPH * FDIM) out[i] = pool[i] / fmaxf(cnt[i >> 6], 1.0f);
}

// ---------------- host orchestration ----------------
extern "C" void kernel_launch(void* const* d_in, const int* in_sizes, int n_in,
                              void* d_out, int out_size, void* d_ws, size_t ws_size,
                              hipStream_t stream) {
  const float* x    = (const float*)d_in[0];
  const int*   ei   = (const int*)d_in[1];      // [2, E] row-major
  const int*   batc = (const int*)d_in[2];
  const float* W1   = (const float*)d_in[3];
  const float* b1   = (const float*)d_in[4];
  const float* W2   = (const float*)d_in[5];
  const float* b2   = (const float*)d_in[6];
  const float* W3   = (const float*)d_in[7];
  const float* b3   = (const float*)d_in[8];
  float* outp = (float*)d_out;

  const int N = NNODES, E = EEDGES;
  const int* src = ei;
  const int* dst = ei + E;

  // workspace carve-out (floats), 256-element aligned
  float* ws = (float*)d_ws;
  size_t off = 0;
  auto alloc = [&](size_t nf) { float* p = ws + off; off += (nf + 255) & ~(size_t)255; return p; };
  float* dinv = alloc((size_t)N);            // deg -> dinv in place
  float* t    = alloc((size_t)N * FDIM);     // GEMM output (pre-scaled by dinv)
  float* hA   = alloc((size_t)N * FDIM);     // layer output ping
  float* hB   = alloc((size_t)N * FDIM);     // layer output pong
  float* pool = alloc((size_t)NGRAPH * FDIM);
  float* cnt  = alloc((size_t)NGRAPH);
  (void)ws_size; (void)n_in; (void)in_sizes; (void)out_size;

  const int ntiles  = N / 16;                // N % 16 == 0
  const int nblkN   = (N + 255) / 256;
  const int nblkE   = (E + 255) / 256;
  const int nblkNF4 = (N * 16 + 255) / 256;  // float4 granularity
  const int gemmBlk = (ntiles + 7) / 8;
  const int nblkEw  = (E + 7) / 8;           // 1 wave per edge, 8 waves/block
  const int nblkNw  = (N + 7) / 8;           // 1 wave per node

  // normalization
  k_deg_init<<<nblkN, 256, 0, stream>>>(dinv, N);
  k_deg_count<<<nblkE, 256, 0, stream>>>(dst, dinv, E);
  k_dinv<<<nblkN, 256, 0, stream>>>(dinv, N);

  // layer 1: x -> hA
  k_gemm_scaled<0><<<gemmBlk, 256, 0, stream>>>(x, W1, dinv, t, ntiles);
  k_selfinit<<<nblkNF4, 256, 0, stream>>>((const float4*)t, dinv, (const float4*)b1, (float4*)hA, N);
  k_scatter<<<nblkEw, 256, 0, stream>>>(src, dst, t, dinv, hA, E);

  // layer 2: relu(hA) -> hB
  k_gemm_scaled<1><<<gemmBlk, 256, 0, stream>>>(hA, W2, dinv, t, ntiles);
  k_selfinit<<<nblkNF4, 256, 0, stream>>>((const float4*)t, dinv, (const float4*)b2, (float4*)hB, N);
  k_scatter<<<nblkEw, 256, 0, stream>>>(src, dst, t, dinv, hB, E);

  // layer 3: relu(hB) -> hA (node embeddings)
  k_gemm_scaled<1><<<gemmBlk, 256, 0, stream>>>(hB, W3, dinv, t, ntiles);
  k_selfinit<<<nblkNF4, 256, 0, stream>>>((const float4*)t, dinv, (const float4*)b3, (float4*)hA, N);
  k_scatter<<<nblkEw, 256, 0, stream>>>(src, dst, t, dinv, hA, E);

  // mean pool -> d_out
  k_pool_zero<<<(NGRAPH * FDIM + 255) / 256, 256, 0, stream>>>(pool, cnt);
  k_pool_acc<<<nblkNw, 256, 0, stream>>>(hA, batc, pool, cnt, N);
  k_pool_out<<<(NGRAPH * FDIM + 255) / 256, 256, 0, stream>>>(pool, cnt, outp);
}